// MLSMPO_43370579755186
// MI455X (gfx1250) — compile-verified
//
#include <hip/hip_runtime.h>
#include <math.h>

// Only check on the device pass: amdgcn builtins are not declared on host.
#if defined(__HIP_DEVICE_COMPILE__)
#if !__has_builtin(__builtin_amdgcn_wmma_f32_16x16x4_f32)
#error "missing __builtin_amdgcn_wmma_f32_16x16x4_f32 on gfx1250 device pass"
#endif
#endif

typedef __attribute__((ext_vector_type(2))) float v2f;
typedef __attribute__((ext_vector_type(8))) float v8f;

#define DD 64
#define VSTR 66      // v-buffer row stride (pad to dodge LDS bank conflicts)
#define MSTR 68      // matrix buffer row stride
#define RELE 1e-6f

// ---------------- block-wide sum over 256 threads (8 waves of 32) -----------
__device__ __forceinline__ float block_sum(float v, volatile float* scr) {
  __syncthreads();
#pragma unroll
  for (int off = 16; off > 0; off >>= 1) v += __shfl_xor(v, off, 32);
  if ((threadIdx.x & 31) == 0) scr[threadIdx.x >> 5] = v;
  __syncthreads();
  float tot = 0.f;
#pragma unroll
  for (int i = 0; i < 8; ++i) tot += scr[i];
  return tot;
}

// ------------- one 16x16 output tile, K=64, fp32 WMMA (16x16x4) -------------
// A: MxK (AT=false: A[m][k] = A[m*lda+k]; AT=true: A[m][k] = A[k*lda+m])
// B: KxN  B[k][n] = B[k*ldb+n]
// Layouts per CDNA5 ISA 7.12.2:
//   A 16x4: lanes 0-15 hold M=lane, VGPR0/1 = K={0,1}; lanes 16-31: K={2,3}
//   B 4x16 mirrored; C/D: VGPR r -> (M = r + 8*(lane>=16), N = lane&15)
template <bool AT>
__device__ __forceinline__ v8f wmma_k64(const float* A, int lda, int m0,
                                        const float* B, int ldb, int n0,
                                        v8f acc) {
  const int lane = threadIdx.x & 31;
  const int li = lane & 15;
  const int kh = (lane >> 4) << 1;  // 0 or 2
#pragma unroll
  for (int kc = 0; kc < 64; kc += 4) {
    const int k0 = kc + kh;
    v2f a, b;
    if (AT) {
      a.x = A[k0 * lda + m0 + li];
      a.y = A[(k0 + 1) * lda + m0 + li];
    } else {
      a.x = A[(m0 + li) * lda + k0];
      a.y = A[(m0 + li) * lda + k0 + 1];
    }
    b.x = B[k0 * ldb + n0 + li];
    b.y = B[(k0 + 1) * ldb + n0 + li];
#if defined(__HIP_DEVICE_COMPILE__)
    acc = __builtin_amdgcn_wmma_f32_16x16x4_f32(false, a, false, b, (short)0,
                                                acc, false, false);
#endif
  }
  return acc;
}

__device__ __forceinline__ void store_tile(float* D, int ldd, int m0, int n0,
                                           v8f acc) {
  const int lane = threadIdx.x & 31;
  const int li = lane & 15, hi = lane >> 4;
#pragma unroll
  for (int r = 0; r < 8; ++r)
    D[(m0 + r + 8 * hi) * ldd + n0 + li] = acc[r];
}

// ------------------- squeezed-input gather (per reference) ------------------
__device__ __forceinline__ float xs_load(const float* __restrict__ src,
                                         int layer, int b, int l, int q) {
  if (layer == 0) {
    // x: (128,32,32,3); l = 16*i + j; q = (di*2+dj)*3 + ch
    int i = l >> 4, j = l & 15;
    int t = q / 3, ch = q - 3 * t;
    int di = t >> 1, dj = t & 1;
    return src[((b * 32 + (2 * i + di)) * 32 + (2 * j + dj)) * 3 + ch];
  } else {
    // h: (128,256) viewed (128,16,16,1); l = 8*i + j; q = di*2+dj
    int i = l >> 3, j = l & 7;
    int di = q >> 1, dj = q & 1;
    return src[b * 256 + (2 * i + di) * 16 + (2 * j + dj)];
  }
}

// ======================= mpo_scale (weights-only scalar) ====================
// One workgroup per layer; rho chain kept in LDS; rho_new tiles accumulate in
// VGPRs across the p-loop via fp32 WMMA.
__device__ void mpo_transfer(float* rho, float* tmp, const float* __restrict__ W,
                             int pm, int wave) {
  const v8f z = {0.f, 0.f, 0.f, 0.f, 0.f, 0.f, 0.f, 0.f};
  v8f acc[2];
  acc[0] = z; acc[1] = z;
  for (int q = 0; q < pm; ++q) {
    // GEMM1: tmp(b,c) = sum_a rho(a,b) * W[(a*pm+q)*64 + c]   (A = rho^T read)
    int nt = 0;
    for (int t = wave; t < 16; t += 8, ++nt) {
      int m0 = (t >> 2) << 4, n0 = (t & 3) << 4;
      v8f ta = wmma_k64<true>(rho, MSTR, m0, W + q * DD, pm * DD, n0, z);
      store_tile(tmp, MSTR, m0, n0, ta);
    }
    __syncthreads();
    // GEMM2: rho_new(c,d) += sum_b tmp(b,c) * W[(b*pm+q)*64 + d]
    nt = 0;
    for (int t = wave; t < 16; t += 8, ++nt) {
      int m0 = (t >> 2) << 4, n0 = (t & 3) << 4;
      acc[nt] = wmma_k64<true>(tmp, MSTR, m0, W + q * DD, pm * DD, n0, acc[nt]);
    }
    __syncthreads();
  }
  int nt = 0;
  for (int t = wave; t < 16; t += 8, ++nt) {
    int m0 = (t >> 2) << 4, n0 = (t & 3) << 4;
    store_tile(rho, MSTR, m0, n0, acc[nt]);
  }
  __syncthreads();
}

__global__ __launch_bounds__(256) void mpo_kernel(
    const float* __restrict__ Wf0, const float* __restrict__ Wm0,
    const float* __restrict__ Wo0, const float* __restrict__ Wl0,
    const float* __restrict__ Wf1, const float* __restrict__ Wm1,
    const float* __restrict__ Wo1, const float* __restrict__ Wl1,
    const float* __restrict__ Wf2, const float* __restrict__ Wm2,
    const float* __restrict__ Wl2, float* __restrict__ cs) {
  __shared__ float rho[DD * MSTR];
  __shared__ float tmp[DD * MSTR];
  __shared__ float scr[8];

  const int layer = blockIdx.x;
  const int tid = threadIdx.x;
  const int wave = tid >> 5;

  const float *Wf, *Wm, *Wo, *Wl;
  int s, p, o, nb;
  if (layer == 0) { Wf = Wf0; Wm = Wm0; Wo = Wo0; Wl = Wl0; s = 32; p = 12; o = 2; nb = 8; }
  else if (layer == 1) { Wf = Wf1; Wm = Wm1; Wo = Wo1; Wl = Wl1; s = 8; p = 4; o = 2; nb = 8; }
  else { Wf = Wf2; Wm = Wm2; Wo = nullptr; Wl = Wl2; s = 64; p = 4; o = 10; nb = 1; }

  // rho = Wf^T Wf  (p x 64 -> 64x64)
#pragma unroll 4
  for (int i2 = 0; i2 < 16; ++i2) {
    int idx = tid + 256 * i2;
    int a = idx >> 6, bb = idx & 63;
    float acc = 0.f;
    for (int q = 0; q < p; ++q) acc += Wf[q * DD + a] * Wf[q * DD + bb];
    rho[a * MSTR + bb] = acc;
  }
  __syncthreads();

  int mid = 0;
  for (int blk = 0; blk < nb; ++blk) {
    int n = blk ? (s - 1) : (s - 2);
    for (int i = 0; i < n; ++i, ++mid)
      mpo_transfer(rho, tmp, Wm + (size_t)mid * DD * p * DD, p, wave);
    if (blk < nb - 1)
      mpo_transfer(rho, tmp, Wo + (size_t)blk * DD * (p * o) * DD, p * o, wave);
  }

  // norm2 = sum_{a,b} rho(a,b) * sum_{q'} Wl[a,q'] Wl[b,q']
  const int po = p * o;
  float lsum = 0.f;
  for (int i2 = 0; i2 < 16; ++i2) {
    int idx = tid + 256 * i2;
    int a = idx >> 6, bb = idx & 63;
    float g = 0.f;
    for (int q = 0; q < po; ++q) g += Wl[a * po + q] * Wl[bb * po + q];
    lsum += rho[a * MSTR + bb] * g;
  }
  float norm2 = block_sum(lsum, scr);
  if (tid == 0) cs[layer] = powf(norm2, -0.5f / (float)(nb * s));
}

// =========================== fused layer_apply ==============================
// One workgroup per batch element. v lives in LDS (rows >= k kept zero so
// partial 16-row tiles are exact). Each site matrix is built on the fly in
// LDS (clamp + Frobenius-normalize), then the fold GEMM runs on fp32 WMMA.
__global__ __launch_bounds__(256) void fold_kernel(
    const float* __restrict__ src, int layer, int L, int s, int p, int o,
    int nb, int prune, const float* __restrict__ Wf,
    const float* __restrict__ Wm, const float* __restrict__ Wo,
    const float* __restrict__ Wl, const float* __restrict__ cs,
    float* __restrict__ dst) {
  __shared__ float sv[128 * VSTR];  // state v (k x 64), zero-padded rows
  __shared__ float sM[DD * MSTR];   // current site matrix (64 x 64)
  __shared__ float sx[16];          // current scaled input vector (p <= 12)
  __shared__ float scr[8];

  const int tid = threadIdx.x;
  const int b = blockIdx.x;
  const int wave = tid >> 5;

  // ---- per-batch normalization: g_b = exp(-mean_l log||xs[b,l,:]||) -------
  float lsum = 0.f;
  for (int l = tid; l < L; l += 256) {
    float ss = 0.f;
    for (int q = 0; q < p; ++q) {
      float x = xs_load(src, layer, b, l, q);
      ss += x * x;
    }
    lsum += 0.5f * logf(ss);
  }
  float tot = block_sum(lsum, scr);
  const float cg = cs[layer] * expf(-tot / (float)L);

  // ---- zero v buffer -------------------------------------------------------
  for (int i = tid; i < 128 * VSTR; i += 256) sv[i] = 0.f;
  __syncthreads();

  // ---- m_first -> v row 0 --------------------------------------------------
  if (tid < p) sx[tid] = cg * xs_load(src, layer, b, 0, tid);
  __syncthreads();
  float f0 = 0.f;
  if (tid < DD) {
    float acc = 0.f;
    for (int q = 0; q < p; ++q) acc += sx[q] * Wf[q * DD + tid];
    if (prune) acc = fmaxf(acc, RELE);
    f0 = acc;
  }
  float n2 = block_sum(f0 * f0, scr);
  if (tid < DD) sv[tid] = prune ? f0 * (1.f / sqrtf(n2)) : f0;
  __syncthreads();

  int k = 1;
  int mid = 0;
  for (int blk = 0; blk < nb; ++blk) {
    const int n = (blk == 0) ? (s - 2) : (s - 1);
    for (int i = 0; i < n; ++i, ++mid) {
      // position of this mid site
      int pos;
      if (mid < s - 2) pos = mid + 1;
      else {
        int r = mid - (s - 2);
        pos = s * (1 + r / (s - 1)) + (r % (s - 1));
      }
      if (tid < p) sx[tid] = cg * xs_load(src, layer, b, pos, tid);
      __syncthreads();

      // build M(d,e) = sum_q sx[q] * Wm[mid,d,q,e]  (clamp + Frob-normalize)
      const float* Wbase = Wm + (size_t)mid * DD * p * DD;
      float vals[16];
      float ssq = 0.f;
#pragma unroll 4
      for (int i2 = 0; i2 < 16; ++i2) {
        int idx = tid + 256 * i2;
        int d = idx >> 6, e = idx & 63;
        float a = 0.f;
        for (int q = 0; q < p; ++q) a += sx[q] * Wbase[(d * p + q) * DD + e];
        if (prune) a = fmaxf(a, RELE);
        vals[i2] = a;
        ssq += a * a;
      }
      float sc = 1.f;
      if (prune) sc = 1.f / sqrtf(block_sum(ssq, scr));
#pragma unroll 4
      for (int i2 = 0; i2 < 16; ++i2) {
        int idx = tid + 256 * i2;
        sM[(idx >> 6) * MSTR + (idx & 63)] = vals[i2] * sc;
      }
      __syncthreads();

      // v <- v @ M  (WMMA fp32, tiles over 8 waves; in-place with barrier)
      const int kt = (k + 15) >> 4;
      const int tiles = kt * 4;
      v8f accs[4];
      int nt = 0;
      for (int t = wave; t < tiles; t += 8, ++nt) {
        int m0 = (t >> 2) << 4, n0 = (t & 3) << 4;
        v8f z = {0.f, 0.f, 0.f, 0.f, 0.f, 0.f, 0.f, 0.f};
        accs[nt] = wmma_k64<false>(sv, VSTR, m0, sM, MSTR, n0, z);
      }
      __syncthreads();
      nt = 0;
      for (int t = wave; t < tiles; t += 8, ++nt) {
        int m0 = (t >> 2) << 4, n0 = (t & 3) << 4;
        store_tile(sv, VSTR, m0, n0, accs[nt]);
      }
      __syncthreads();
    }

    if (blk < nb - 1) {  // out site: M_out (64 x 2 x 64), k doubles (o==2)
      const int pos = (blk + 1) * s - 1;
      if (tid < p) sx[tid] = cg * xs_load(src, layer, b, pos, tid);
      __syncthreads();
      const float* Wbase = Wo + (size_t)blk * DD * p * 2 * DD;

      // pass 1: sum of squares over the full clamped (d,o,e) tensor
      float ssq = 0.f;
      for (int i2 = 0; i2 < 32; ++i2) {
        int idx = tid + 256 * i2;  // 8192 entries
        int d = idx >> 7, rem = idx & 127, oo = rem >> 6, e = rem & 63;
        float a = 0.f;
        for (int q = 0; q < p; ++q)
          a += sx[q] * Wbase[((d * p + q) * 2 + oo) * DD + e];
        a = fmaxf(a, RELE);
        ssq += a * a;
      }
      const float sc = 1.f / sqrtf(block_sum(ssq, scr));

      const int kt = (k + 15) >> 4;  // k <= 64 here -> tiles <= 16
      const int tiles = kt * 4;
      v8f oacc[2][2];
      for (int h = 0; h < 2; ++h) {
        // rebuild half h of M_out into sM
#pragma unroll 4
        for (int i2 = 0; i2 < 16; ++i2) {
          int idx = tid + 256 * i2;
          int d = idx >> 6, e = idx & 63;
          float a = 0.f;
          for (int q = 0; q < p; ++q)
            a += sx[q] * Wbase[((d * p + q) * 2 + h) * DD + e];
          sM[d * MSTR + e] = fmaxf(a, RELE) * sc;
        }
        __syncthreads();
        int nt = 0;
        for (int t = wave; t < tiles; t += 8, ++nt) {
          int m0 = (t >> 2) << 4, n0 = (t & 3) << 4;
          v8f z = {0.f, 0.f, 0.f, 0.f, 0.f, 0.f, 0.f, 0.f};
          oacc[h][nt] = wmma_k64<false>(sv, VSTR, m0, sM, MSTR, n0, z);
        }
        __syncthreads();
      }
      // interleaved write-back: new row = 2*row + h  (matches reshape)
      {
        const int lane = tid & 31, li = lane & 15, hi = lane >> 4;
        for (int h = 0; h < 2; ++h) {
          int nt = 0;
          for (int t = wave; t < tiles; t += 8, ++nt) {
            int m0 = (t >> 2) << 4, n0 = (t & 3) << 4;
#pragma unroll
            for (int r = 0; r < 8; ++r) {
              int row = m0 + r + 8 * hi;
              sv[(2 * row + h) * VSTR + n0 + li] = oacc[h][nt][r];
            }
          }
        }
      }
      __syncthreads();
      k <<= 1;
    }
  }

  // ---- m_last + final contraction -----------------------------------------
  if (tid < p) sx[tid] = cg * xs_load(src, layer, b, L - 1, tid);
  __syncthreads();
  const int no = DD * o;  // 128 (o=2) or 640 (o=10)
  float mssq = 0.f;
  for (int idx = tid; idx < no; idx += 256) {
    int d = idx / o, oo = idx - d * o;
    float a = 0.f;
    for (int q = 0; q < p; ++q) a += sx[q] * Wl[(d * p + q) * o + oo];
    if (prune) a = fmaxf(a, RELE);
    sM[idx] = a;
    mssq += a * a;
  }
  float msum = block_sum(mssq, scr);
  if (prune) {
    float msc = 1.f / sqrtf(msum);
    for (int idx = tid; idx < no; idx += 256) sM[idx] *= msc;
  }
  __syncthreads();

  if (o == 2) {  // layers 0,1: k == 128 -> 256 outputs per batch element
    int kk = tid >> 1, oo = tid & 1;
    float a = 0.f;
    for (int d = 0; d < DD; ++d) a += sv[kk * VSTR + d] * sM[d * 2 + oo];
    dst[b * 256 + tid] = a;
  } else {  // layer 2: k == 1, o == 10; clamp + L2-normalize the row
    if (tid < o) {
      float a = 0.f;
      for (int d = 0; d < DD; ++d) a += sv[d] * sM[d * o + tid];
      sx[tid] = fmaxf(a, RELE);
    }
    __syncthreads();
    if (tid < o) {
      float ss = 0.f;
      for (int j = 0; j < o; ++j) ss += sx[j] * sx[j];
      dst[b * o + tid] = sx[tid] * (1.f / sqrtf(ss));
    }
  }
}

// ============================== batchnorm ===================================
__global__ __launch_bounds__(256) void bn_kernel(float* __restrict__ h,
                                                 const float* __restrict__ g,
                                                 const float* __restrict__ bt) {
  const int j = threadIdx.x;  // 256 features, thread-private column -> no races
  float mu = 0.f;
  for (int b = 0; b < 128; ++b) mu += h[b * 256 + j];
  mu *= (1.f / 128.f);
  float var = 0.f;
  for (int b = 0; b < 128; ++b) {
    float d = h[b * 256 + j] - mu;
    var += d * d;
  }
  var *= (1.f / 128.f);
  const float inv = rsqrtf(var + 1e-5f);
  const float gg = g[j], bb = bt[j];
  for (int b = 0; b < 128; ++b)
    h[b * 256 + j] = gg * (h[b * 256 + j] - mu) * inv + bb;
}

// ============================== launcher ====================================
extern "C" void kernel_launch(void* const* d_in, const int* in_sizes, int n_in,
                              void* d_out, int out_size, void* d_ws,
                              size_t ws_size, hipStream_t stream) {
  (void)in_sizes; (void)n_in; (void)out_size; (void)ws_size;
  const float* x   = (const float*)d_in[0];
  const float* Wf0 = (const float*)d_in[1];
  const float* Wm0 = (const float*)d_in[2];
  const float* Wo0 = (const float*)d_in[3];
  const float* Wl0 = (const float*)d_in[4];
  const float* g0  = (const float*)d_in[5];
  const float* b0  = (const float*)d_in[6];
  const float* Wf1 = (const float*)d_in[7];
  const float* Wm1 = (const float*)d_in[8];
  const float* Wo1 = (const float*)d_in[9];
  const float* Wl1 = (const float*)d_in[10];
  const float* g1  = (const float*)d_in[11];
  const float* b1  = (const float*)d_in[12];
  const float* Wf2 = (const float*)d_in[13];
  const float* Wm2 = (const float*)d_in[14];
  // d_in[15] = Wo2 is empty (nb==1), never dereferenced
  const float* Wl2 = (const float*)d_in[16];

  float* cs = (float*)d_ws;        // 3 scalars (padded)
  float* h0 = cs + 64;             // 128 x 256
  float* h1 = h0 + 128 * 256;      // 128 x 256
  float* out = (float*)d_out;      // 128 x 10

  mpo_kernel<<<3, 256, 0, stream>>>(Wf0, Wm0, Wo0, Wl0, Wf1, Wm1, Wo1, Wl1,
                                    Wf2, Wm2, Wl2, cs);
  fold_kernel<<<128, 256, 0, stream>>>(x, 0, 256, 32, 12, 2, 8, 1, Wf0, Wm0,
                                       Wo0, Wl0, cs, h0);
  bn_kernel<<<1, 256, 0, stream>>>(h0, g0, b0);
  fold_kernel<<<128, 256, 0, stream>>>(h0, 1, 64, 8, 4, 2, 8, 1, Wf1, Wm1,
                                       Wo1, Wl1, cs, h1);
  bn_kernel<<<1, 256, 0, stream>>>(h1, g1, b1);
  fold_kernel<<<128, 256, 0, stream>>>(h1, 2, 64, 64, 4, 10, 1, 0, Wf2, Wm2,
                                       nullptr, Wl2, cs, out);
}